// MultiHeadAttention_37623913513848
// MI455X (gfx1250) — compile-verified
//
#include <hip/hip_runtime.h>
#include <math.h>

#define T_SEQ 2048
#define CDIM  1024
#define NHEAD 16
#define HDIM  64
#define BATCH 2

typedef __attribute__((ext_vector_type(2))) float v2f;
typedef __attribute__((ext_vector_type(8))) float v8f;
typedef __attribute__((ext_vector_type(4))) int  v4i;

#if defined(__gfx1250__) && __has_builtin(__builtin_amdgcn_global_load_async_to_lds_b128)
#define ASYNC_LDS 1
typedef const __attribute__((address_space(1))) void* gas_cvptr;
typedef __attribute__((address_space(3))) void*       las_vptr;
typedef __attribute__((address_space(1))) v4i*        g_v4i_ptr;
typedef __attribute__((address_space(3))) v4i*        l_v4i_ptr;
#else
#define ASYNC_LDS 0
#endif

// 16B global -> LDS tile copy (async DMA path on CDNA5 when available)
__device__ __forceinline__ void copy16(float* lds, const float* g) {
#if ASYNC_LDS
  g_v4i_ptr gp = (g_v4i_ptr)(gas_cvptr)g;   // addrspacecast, then retype/de-const
  l_v4i_ptr lp = (l_v4i_ptr)(las_vptr)lds;
  __builtin_amdgcn_global_load_async_to_lds_b128(gp, lp, 0, 0);
#else
  *reinterpret_cast<float4*>(lds) = *reinterpret_cast<const float4*>(g);
#endif
}
__device__ __forceinline__ void copy_fence() {
#if ASYNC_LDS
#if __has_builtin(__builtin_amdgcn_s_wait_asynccnt)
  __builtin_amdgcn_s_wait_asynccnt(0);
#else
  asm volatile("s_wait_asynccnt 0" ::: "memory");
#endif
#endif
}

__device__ __forceinline__ v8f wmma_f32(v2f a, v2f b, v8f c) {
  // D = A(16x4) * B(4x16) + C, fp32 end to end
  return __builtin_amdgcn_wmma_f32_16x16x4_f32(false, a, false, b, (short)0, c, false, false);
}

__device__ __forceinline__ float half16_max(float v) {
#pragma unroll
  for (int m = 1; m < 16; m <<= 1) v = fmaxf(v, __shfl_xor(v, m, 32));
  return v;
}
__device__ __forceinline__ float half16_sum(float v) {
#pragma unroll
  for (int m = 1; m < 16; m <<= 1) v += __shfl_xor(v, m, 32);
  return v;
}

// ---------------- WMMA f32 GEMM: out = A[M,K] @ Bmat[K,N] ----------------
// Double-buffered LDS staging (async prefetch of slab k+1 overlaps WMMAs on slab k).
// MODE 0: scatter into Q/K/V as [B,H,T,D].  MODE 1: +bias, row-major store.
template <int MODE>
__global__ void __launch_bounds__(256)
gemm_wmma_f32(const float* __restrict__ A, const float* __restrict__ Bmat,
              const float* __restrict__ bias, float* __restrict__ outQ,
              float* __restrict__ outK, float* __restrict__ outV,
              int M, int N, int K) {
  __shared__ __align__(16) float As[2][128 * 20];  // 128 rows x 16 k, stride 20
  __shared__ __align__(16) float Bs[2][16 * 68];   // 16 k x 64 n, stride 68
  const int tid  = threadIdx.x;
  const int wave = tid >> 5, lane = tid & 31;
  const int li = lane & 15, lh = lane >> 4;
  const int wm = wave >> 1, wn = wave & 1;         // 4x2 wave grid, 32x32 per wave
  const int m_blk = blockIdx.x * 128;
  const int n_blk = blockIdx.y * 64;

  auto stageAB = [&](int buf, int k0) {
#pragma unroll
    for (int j = 0; j < 2; ++j) {                  // A slab: 128x16 floats
      int f = tid + j * 256;
      int r = f >> 2, c4 = (f & 3) << 2;
      copy16(&As[buf][r * 20 + c4], A + (size_t)(m_blk + r) * K + k0 + c4);
    }
    {                                              // B slab: 16x64 floats
      int r = tid >> 4, c4 = (tid & 15) << 2;
      copy16(&Bs[buf][r * 68 + c4], Bmat + (size_t)(k0 + r) * N + n_blk + c4);
    }
  };

  v8f acc[2][2];
#pragma unroll
  for (int i = 0; i < 2; ++i)
#pragma unroll
    for (int j = 0; j < 2; ++j)
#pragma unroll
      for (int e = 0; e < 8; ++e) acc[i][j][e] = 0.0f;

  stageAB(0, 0);
  copy_fence();
  __syncthreads();

  const int nk = K >> 4;
  for (int kk = 0; kk < nk; ++kk) {
    const int buf = kk & 1;
    if (kk + 1 < nk) stageAB(buf ^ 1, (kk + 1) << 4);  // async prefetch next slab
    const float* Asb = &As[buf][0];
    const float* Bsb = &Bs[buf][0];
#pragma unroll
    for (int dd = 0; dd < 4; ++dd) {               // 4 k-steps of 4
      v2f af[2], bf[2];
#pragma unroll
      for (int sm = 0; sm < 2; ++sm)
        af[sm] = *reinterpret_cast<const v2f*>(
            &Asb[(wm * 32 + sm * 16 + li) * 20 + dd * 4 + 2 * lh]);
#pragma unroll
      for (int sn = 0; sn < 2; ++sn) {
        v2f b;
        b.x = Bsb[(dd * 4 + 2 * lh + 0) * 68 + wn * 32 + sn * 16 + li];
        b.y = Bsb[(dd * 4 + 2 * lh + 1) * 68 + wn * 32 + sn * 16 + li];
        bf[sn] = b;
      }
#pragma unroll
      for (int sm = 0; sm < 2; ++sm)
#pragma unroll
        for (int sn = 0; sn < 2; ++sn)
          acc[sm][sn] = wmma_f32(af[sm], bf[sn], acc[sm][sn]);
    }
    copy_fence();                                   // prefetch writes visible
    __syncthreads();
  }

#pragma unroll
  for (int sm = 0; sm < 2; ++sm)
#pragma unroll
    for (int sn = 0; sn < 2; ++sn) {
      const int n16 = n_blk + wn * 32 + sn * 16;
#pragma unroll
      for (int r = 0; r < 8; ++r) {
        const int m = m_blk + wm * 32 + sm * 16 + r + 8 * lh;
        const float val = acc[sm][sn][r];
        if (MODE == 1) {
          const int n = n16 + li;
          outQ[(size_t)m * N + n] = val + bias[n];
        } else {
          const int sect = n16 >> 10;              // 0=q 1=k 2=v
          const int cw = n16 & 1023;
          const int hh = cw >> 6;
          const int d = (cw & 63) + li;
          const int b = m >> 11, t = m & 2047;
          float* base = (sect == 0) ? outQ : ((sect == 1) ? outK : outV);
          base[(((size_t)b * NHEAD + hh) * T_SEQ + t) * HDIM + d] = val;
        }
      }
    }
}

// ---------------- RoPE in place on Q and K [B,H,T,D] ----------------
__global__ void __launch_bounds__(256)
rope_inplace(float* __restrict__ Q, float* __restrict__ K) {
  const size_t idx = (size_t)blockIdx.x * blockDim.x + threadIdx.x;
  const size_t per = (size_t)BATCH * NHEAD * T_SEQ * 32;  // pairs per buffer
  float* base = (idx < per) ? Q : K;
  const size_t r = (idx < per) ? idx : idx - per;
  const int i = (int)(r & 31);
  const int t = (int)((r >> 5) & 2047);
  const size_t bh = r >> 16;                       // r / (32*2048)
  float* p = base + (bh * T_SEQ + t) * HDIM + i;
  // freq = 10000^(-2i/64); ln(10000)/32 = 0.287823136...
  const float ang = (float)t * __expf(-0.28782313662425576f * (float)i);
  float sn, cs;
  sincosf(ang, &sn, &cs);                          // accurate: ang up to ~2047 rad
  const float lo = p[0], hi = p[32];
  p[0]  = lo * cs - hi * sn;
  p[32] = hi * cs + lo * sn;
}

// ---------------- Causal flash attention, one WG per 64 q rows ----------------
__global__ void __launch_bounds__(128)
attn_flash_wmma(const float* __restrict__ Q, const float* __restrict__ K,
                const float* __restrict__ V, float* __restrict__ Y) {
  __shared__ __align__(16) float Ks[2][64 * 68];
  __shared__ __align__(16) float Vs[2][64 * 68];
  __shared__ __align__(16) float Ps[4][16 * 18];   // per-wave P transpose scratch
  const int tid  = threadIdx.x;
  const int wave = tid >> 5, lane = tid & 31;
  const int li = lane & 15, lh = lane >> 4;
  const int q0 = blockIdx.x * 64;
  const int bh = blockIdx.y;
  const int b = bh >> 4, h = bh & 15;
  const float* Qb = Q + (size_t)bh * T_SEQ * HDIM;
  const float* Kb = K + (size_t)bh * T_SEQ * HDIM;
  const float* Vb = V + (size_t)bh * T_SEQ * HDIM;

  auto stageKV = [&](int buf, int kb) {
#pragma unroll
    for (int j = 0; j < 8; ++j) {                  // 64x64 K and V tiles
      int f = tid + j * 128;
      int r = f >> 4, c4 = (f & 15) << 2;
      copy16(&Ks[buf][r * 68 + c4], Kb + (size_t)(kb + r) * HDIM + c4);
      copy16(&Vs[buf][r * 68 + c4], Vb + (size_t)(kb + r) * HDIM + c4);
    }
  };

  // Q fragments in registers, pre-scaled by 1/sqrt(64)
  v2f qa[16];
  {
    const int qrow = q0 + wave * 16 + li;
#pragma unroll
    for (int dd = 0; dd < 16; ++dd) {
      v2f t = *reinterpret_cast<const v2f*>(Qb + (size_t)qrow * HDIM + dd * 4 + 2 * lh);
      t.x *= 0.125f; t.y *= 0.125f;
      qa[dd] = t;
    }
  }

  v8f o[4];
#pragma unroll
  for (int dt = 0; dt < 4; ++dt)
#pragma unroll
    for (int e = 0; e < 8; ++e) o[dt][e] = 0.0f;
  float mrow[8], lrow[8];
#pragma unroll
  for (int r = 0; r < 8; ++r) { mrow[r] = -3.0e38f; lrow[r] = 0.0f; }

  const int nkb = blockIdx.x + 1;                  // causal: blocks 0..q0/64
  stageKV(0, 0);
  copy_fence();
  __syncthreads();

  for (int ib = 0; ib < nkb; ++ib) {
    const int kb = ib * 64;
    const int buf = ib & 1;
    if (ib + 1 < nkb) stageKV(buf ^ 1, kb + 64);   // async prefetch next K/V tile
    const float* Ksb = &Ks[buf][0];
    const float* Vsb = &Vs[buf][0];

    v8f s[4];                                      // S = Qs @ K^T, 4 tiles of 16x16
#pragma unroll
    for (int n = 0; n < 4; ++n) {
#pragma unroll
      for (int e = 0; e < 8; ++e) s[n][e] = 0.0f;
#pragma unroll
      for (int dd = 0; dd < 16; ++dd) {
        v2f kf = *reinterpret_cast<const v2f*>(&Ksb[(n * 16 + li) * 68 + dd * 4 + 2 * lh]);
        s[n] = wmma_f32(qa[dd], kf, s[n]);
      }
    }

    if (kb == q0) {                                // causal mask on diagonal block
#pragma unroll
      for (int n = 0; n < 4; ++n)
#pragma unroll
        for (int r = 0; r < 8; ++r) {
          const int kcol = kb + n * 16 + li;
          const int qg = q0 + wave * 16 + r + 8 * lh;
          if (kcol > qg) s[n][r] = -3.0e38f;
        }
    }

    // online softmax (row = across lanes within a 16-lane half)
    float scale[8], mnew[8];
#pragma unroll
    for (int r = 0; r < 8; ++r) {
      float rm = fmaxf(fmaxf(s[0][r], s[1][r]), fmaxf(s[2][r], s[3][r]));
      rm = half16_max(rm);
      mnew[r] = fmaxf(mrow[r], rm);
      scale[r] = __expf(mrow[r] - mnew[r]);
    }
#pragma unroll
    for (int r = 0; r < 8; ++r) {
      float rs = 0.0f;
#pragma unroll
      for (int n = 0; n < 4; ++n) {
        float p = __expf(s[n][r] - mnew[r]);
        s[n][r] = p;
        rs += p;
      }
      rs = half16_sum(rs);
      lrow[r] = lrow[r] * scale[r] + rs;
      mrow[r] = mnew[r];
    }
#pragma unroll
    for (int dt = 0; dt < 4; ++dt)
#pragma unroll
      for (int r = 0; r < 8; ++r) o[dt][r] *= scale[r];

    // O += P @ V  (C-layout -> A-layout via per-wave LDS scratch)
#pragma unroll
    for (int n = 0; n < 4; ++n) {
#pragma unroll
      for (int r = 0; r < 8; ++r) Ps[wave][(r + 8 * lh) * 18 + li] = s[n][r];
#pragma unroll
      for (int dt = 0; dt < 4; ++dt)
#pragma unroll
        for (int ss = 0; ss < 4; ++ss) {
          v2f pf = *reinterpret_cast<const v2f*>(&Ps[wave][li * 18 + ss * 4 + 2 * lh]);
          v2f vf;
          vf.x = Vsb[(n * 16 + ss * 4 + 2 * lh + 0) * 68 + dt * 16 + li];
          vf.y = Vsb[(n * 16 + ss * 4 + 2 * lh + 1) * 68 + dt * 16 + li];
          o[dt] = wmma_f32(pf, vf, o[dt]);
        }
    }
    copy_fence();                                  // prefetch writes visible
    __syncthreads();
  }

  // Y[b, t, h*64 + d] = O / l
#pragma unroll
  for (int dt = 0; dt < 4; ++dt)
#pragma unroll
    for (int r = 0; r < 8; ++r) {
      const int qg = q0 + wave * 16 + r + 8 * lh;
      const int d = dt * 16 + li;
      Y[((size_t)b * T_SEQ + qg) * CDIM + h * HDIM + d] = o[dt][r] / lrow[r];
    }
}

extern "C" void kernel_launch(void* const* d_in, const int* in_sizes, int n_in,
                              void* d_out, int out_size, void* d_ws, size_t ws_size,
                              hipStream_t stream) {
  const float* x      = (const float*)d_in[0];  // [2,2048,1024]
  const float* w_qkv  = (const float*)d_in[1];  // [1024,3072]
  const float* w_proj = (const float*)d_in[2];  // [1024,1024]
  const float* b_proj = (const float*)d_in[3];  // [1024]
  float* out = (float*)d_out;

  const size_t elems = (size_t)BATCH * NHEAD * T_SEQ * HDIM;  // 4 Mi floats
  float* Qw = (float*)d_ws;
  float* Kw = Qw + elems;
  float* Vw = Kw + elems;
  float* Yw = Vw + elems;                                     // total 64 MB scratch

  gemm_wmma_f32<0><<<dim3(32, 48), 256, 0, stream>>>(x, w_qkv, nullptr,
                                                     Qw, Kw, Vw, 4096, 3072, 1024);
  rope_inplace<<<16384, 256, 0, stream>>>(Qw, Kw);
  attn_flash_wmma<<<dim3(32, 32), 128, 0, stream>>>(Qw, Kw, Vw, Yw);
  gemm_wmma_f32<1><<<dim3(32, 16), 256, 0, stream>>>(Yw, w_proj, b_proj,
                                                     out, nullptr, nullptr, 4096, 1024, 1024);
}